// SoftmaxTransformer_65876208386315
// MI455X (gfx1250) — compile-verified
//
#include <hip/hip_runtime.h>
#include <math.h>

// SoftmaxTransformer ensemble attention for MI455X (gfx1250).
// All contractions lowered to V_WMMA_F32_16X16X4_F32 (wave32, 16x16 f32 tiles).

typedef float v2f __attribute__((ext_vector_type(2)));
typedef float v8f __attribute__((ext_vector_type(8)));

#define B_   4
#define E_   50
#define C_   64
#define S_   2048           // H*W = 32*64
#define NH_  64
#define NM_  64
#define CHW_ (C_ * S_)      // 131072
#define NX_  (B_ * E_ * C_ * S_)  // 26214400 floats per full tensor

__device__ __forceinline__ v8f wmma4(v2f a, v2f b, v8f c) {
    // D = A(16x4) * B(4x16) + C(16x16), all f32
    return __builtin_amdgcn_wmma_f32_16x16x4_f32(
        /*neg_a=*/false, a, /*neg_b=*/false, b,
        /*c_mod=*/(short)0, c, /*reuse_a=*/false, /*reuse_b=*/false);
}

// ---------------------------------------------------------------- LayerNorm
__global__ void ln_kernel(const float* __restrict__ x, const float* __restrict__ g,
                          const float* __restrict__ bta, float* __restrict__ pre) {
    __shared__ float red0[256];
    __shared__ float red1[256];
    const int be = blockIdx.x;
    const float* xb = x + (size_t)be * CHW_;
    float s = 0.f, s2 = 0.f;
    for (int i = threadIdx.x; i < CHW_; i += 256) {
        float v = xb[i];
        s += v; s2 += v * v;
    }
    red0[threadIdx.x] = s; red1[threadIdx.x] = s2;
    __syncthreads();
    for (int off = 128; off > 0; off >>= 1) {
        if (threadIdx.x < off) {
            red0[threadIdx.x] += red0[threadIdx.x + off];
            red1[threadIdx.x] += red1[threadIdx.x + off];
        }
        __syncthreads();
    }
    const float mean = red0[0] * (1.f / CHW_);
    const float var  = red1[0] * (1.f / CHW_) - mean * mean;
    const float rs   = rsqrtf(var + 1e-5f);
    float* pb = pre + (size_t)be * CHW_;
    for (int i = threadIdx.x; i < CHW_; i += 256)
        pb[i] = (xb[i] - mean) * rs * g[i] + bta[i];
}

// --------------------------------------------- generic 64-K WMMA tile helper
// acc(16x16) = W[m0:m0+16, 0:64] * X[0:64, n0:n0+16]   (X row stride = S_)
__device__ __forceinline__ v8f gemm64_tile(const float* __restrict__ Wm, int m0,
                                           const float* __restrict__ X, int n0) {
    const int lane = threadIdx.x & 31;
    const int lm   = lane & 15;
    const int half = lane >> 4;
    const int arow = (m0 + lm) * 64;
    v8f acc = {};
#pragma unroll
    for (int k0 = 0; k0 < 64; k0 += 4) {
        const int ka = k0 + 2 * half;
        v2f a, b;
        a.x = Wm[arow + ka];
        a.y = Wm[arow + ka + 1];
        b.x = X[ka * S_ + n0 + lm];
        b.y = X[(ka + 1) * S_ + n0 + lm];
        acc = wmma4(a, b, acc);
    }
    return acc;
}

// ------------------------------------------------------------- QKV projection
// v/k/q[b,e,h,s] = sum_c W[h,c] * pre[b,e,c,s]; B-operand shared across 3 GEMMs
__global__ void qkv_kernel(const float* __restrict__ pre,
                           const float* __restrict__ Wv, const float* __restrict__ Wk,
                           const float* __restrict__ Wq,
                           float* __restrict__ v, float* __restrict__ k,
                           float* __restrict__ q) {
    const int wave = blockIdx.x * 8 + (threadIdx.x >> 5);
    const int be = wave >> 9;         // /512
    const int t  = wave & 511;
    const int m0 = (t >> 7) << 4;
    const int n0 = (t & 127) << 4;
    const int lane = threadIdx.x & 31;
    const int lm = lane & 15;
    const int half = lane >> 4;
    const float* X = pre + (size_t)be * CHW_;
    const int arow = (m0 + lm) * 64;
    v8f cv = {}, ck = {}, cq = {};
#pragma unroll
    for (int k0 = 0; k0 < 64; k0 += 4) {
        const int ka = k0 + 2 * half;
        v2f bv, av, ak, aq;
        bv.x = X[ka * S_ + n0 + lm];
        bv.y = X[(ka + 1) * S_ + n0 + lm];
        av.x = Wv[arow + ka]; av.y = Wv[arow + ka + 1];
        ak.x = Wk[arow + ka]; ak.y = Wk[arow + ka + 1];
        aq.x = Wq[arow + ka]; aq.y = Wq[arow + ka + 1];
        cv = wmma4(av, bv, cv);
        ck = wmma4(ak, bv, ck);
        cq = wmma4(aq, bv, cq);
    }
    const int col = n0 + lm;
    const int mb  = m0 + half * 8;
    float* vb = v + (size_t)be * (NH_ * S_);
    float* kb = k + (size_t)be * (NH_ * S_);
    float* qb = q + (size_t)be * (NH_ * S_);
#pragma unroll
    for (int r = 0; r < 8; ++r) {
        vb[(mb + r) * S_ + col] = cv[r];
        kb[(mb + r) * S_ + col] = ck[r];
        qb[(mb + r) * S_ + col] = cq[r];
    }
}

// ----------------------------------------------------------------- Gram k.qT
// gram[b,h,i,j] = scale * sum_s k[b,i,h,s]*q[b,j,h,s]; E padded to 64
__global__ void gram_kernel(const float* __restrict__ k, const float* __restrict__ q,
                            float* __restrict__ gram) {
    const int wave = blockIdx.x * 8 + (threadIdx.x >> 5);
    const int bh = wave >> 4;          // 0..255
    const int t  = wave & 15;
    const int m0 = (t >> 2) << 4;      // i tile
    const int n0 = (t & 3) << 4;       // j tile
    const int b = bh >> 6, h = bh & 63;
    const int lane = threadIdx.x & 31;
    const int lm = lane & 15;
    const int half = lane >> 4;
    const int i = m0 + lm, j = n0 + lm;
    const bool iv = i < E_, jv = j < E_;
    const size_t istr = (size_t)NH_ * S_;
    const float* kp = k + ((size_t)b * E_ * NH_ + h) * S_ + (iv ? (size_t)i * istr : 0);
    const float* qp = q + ((size_t)b * E_ * NH_ + h) * S_ + (jv ? (size_t)j * istr : 0);
    v8f acc = {};
#pragma unroll 4
    for (int s0 = 0; s0 < S_; s0 += 4) {
        const int sa = s0 + 2 * half;
        v2f a, bb;
        a.x  = iv ? kp[sa]     : 0.f;
        a.y  = iv ? kp[sa + 1] : 0.f;
        bb.x = jv ? qp[sa]     : 0.f;
        bb.y = jv ? qp[sa + 1] : 0.f;
        acc = wmma4(a, bb, acc);
    }
    const float scale = 0.022097086912079608f; // 1/sqrt(2048)
    const int col = n0 + lm;
    const int mb  = m0 + half * 8;
    float* gp = gram + (size_t)bh * (E_ * E_);
#pragma unroll
    for (int r = 0; r < 8; ++r) {
        const int mi = mb + r;
        if (mi < E_ && col < E_) gp[mi * E_ + col] = acc[r] * scale;
    }
}

// ------------------------------------------------- softmax over key axis (i)
__global__ void softmax_kernel(float* __restrict__ gram) {
    const int bh = blockIdx.x;
    const int j = threadIdx.x;
    if (j >= E_) return;
    float* g = gram + (size_t)bh * (E_ * E_);
    float mx = -1e30f;
    for (int i = 0; i < E_; ++i) mx = fmaxf(mx, g[i * E_ + j]);
    float sum = 0.f;
    for (int i = 0; i < E_; ++i) sum += expf(g[i * E_ + j] - mx);
    const float inv = 1.f / sum;
    for (int i = 0; i < E_; ++i) g[i * E_ + j] = expf(g[i * E_ + j] - mx) * inv;
}

// -------------------------------------------------------- ensemble mean of v
__global__ void vmean_kernel(const float* __restrict__ v, float* __restrict__ vm) {
    const int idx = blockIdx.x * 256 + threadIdx.x;  // over B*NH*S
    const int b = idx / (NH_ * S_);
    const int hs = idx % (NH_ * S_);
    const float* vp = v + (size_t)b * E_ * NH_ * S_ + hs;
    float s = 0.f;
    for (int e = 0; e < E_; ++e) s += vp[(size_t)e * NH_ * S_];
    vm[idx] = s * (1.f / E_);
}

// ---------------------------------------- transformed = v + w^T (v - vmean)
__global__ void mix_kernel(const float* __restrict__ v, const float* __restrict__ vm,
                           const float* __restrict__ w, float* __restrict__ tout) {
    const int wave = blockIdx.x * 8 + (threadIdx.x >> 5);
    const int bh = wave >> 9;          // 0..255
    const int t  = wave & 511;
    const int m0 = (t >> 7) << 4;      // j tile
    const int n0 = (t & 127) << 4;     // s tile
    const int b = bh >> 6, h = bh & 63;
    const int lane = threadIdx.x & 31;
    const int lm = lane & 15;
    const int half = lane >> 4;
    const int j = m0 + lm;
    const bool jv = j < E_;
    const float* wp = w + (size_t)bh * (E_ * E_);               // w[i*E + j]
    const float* vbase = v + ((size_t)b * E_ * NH_ + h) * S_;   // + i*NH*S
    const float* vmp = vm + ((size_t)b * NH_ + h) * S_;
    const size_t istr = (size_t)NH_ * S_;
    const int col = n0 + lm;
    const float vm0 = vmp[col];
    v8f acc = {};
#pragma unroll
    for (int i0 = 0; i0 < 52; i0 += 4) {                        // K=E padded
        const int ia = i0 + 2 * half;
        v2f a, bb;
        a.x  = (jv && ia     < E_) ? wp[ia * E_ + j]       : 0.f;
        a.y  = (jv && ia + 1 < E_) ? wp[(ia + 1) * E_ + j] : 0.f;
        bb.x = (ia     < E_) ? (vbase[(size_t)ia * istr + col] - vm0)       : 0.f;
        bb.y = (ia + 1 < E_) ? (vbase[(size_t)(ia + 1) * istr + col] - vm0) : 0.f;
        acc = wmma4(a, bb, acc);
    }
    const int mbj = m0 + half * 8;
    float* tb = tout + ((size_t)b * E_ * NH_ + h) * S_;
#pragma unroll
    for (int r = 0; r < 8; ++r) {
        const int jj = mbj + r;
        if (jj < E_)
            tb[(size_t)jj * istr + col] = vbase[(size_t)jj * istr + col] + acc[r];
    }
}

// ---------------------------------------------- after = x + W_out @ transformed
__global__ void wout_kernel(const float* __restrict__ tin, const float* __restrict__ Wout,
                            const float* __restrict__ x, float* __restrict__ after) {
    const int wave = blockIdx.x * 8 + (threadIdx.x >> 5);
    const int be = wave >> 9;
    const int t  = wave & 511;
    const int m0 = (t >> 7) << 4;
    const int n0 = (t & 127) << 4;
    v8f acc = gemm64_tile(Wout, m0, tin + (size_t)be * (NH_ * S_), n0);
    const int lane = threadIdx.x & 31;
    const int col = n0 + (lane & 15);
    const int mb  = m0 + (lane >> 4) * 8;
    const float* xb = x + (size_t)be * CHW_;
    float* ab = after + (size_t)be * CHW_;
#pragma unroll
    for (int r = 0; r < 8; ++r)
        ab[(mb + r) * S_ + col] = xb[(mb + r) * S_ + col] + acc[r];
}

// ---------------------------------- h1 = gelu_exact(W_m1 @ after + b_m1)
__global__ void m1_kernel(const float* __restrict__ after, const float* __restrict__ Wm1,
                          const float* __restrict__ b1, float* __restrict__ h1) {
    const int wave = blockIdx.x * 8 + (threadIdx.x >> 5);
    const int be = wave >> 9;
    const int t  = wave & 511;
    const int m0 = (t >> 7) << 4;
    const int n0 = (t & 127) << 4;
    v8f acc = gemm64_tile(Wm1, m0, after + (size_t)be * CHW_, n0);
    const int lane = threadIdx.x & 31;
    const int col = n0 + (lane & 15);
    const int mb  = m0 + (lane >> 4) * 8;
    float* hb = h1 + (size_t)be * (NM_ * S_);
#pragma unroll
    for (int r = 0; r < 8; ++r) {
        const float u = acc[r] + b1[mb + r];
        hb[(mb + r) * S_ + col] = 0.5f * u * (1.f + erff(u * 0.70710678118654752f));
    }
}

// ------------------------------------ out = after + W_m2 @ h1 + b_m2
__global__ void m2_kernel(const float* __restrict__ h1, const float* __restrict__ Wm2,
                          const float* __restrict__ b2, const float* __restrict__ after,
                          float* __restrict__ out) {
    const int wave = blockIdx.x * 8 + (threadIdx.x >> 5);
    const int be = wave >> 9;
    const int t  = wave & 511;
    const int m0 = (t >> 7) << 4;
    const int n0 = (t & 127) << 4;
    v8f acc = gemm64_tile(Wm2, m0, h1 + (size_t)be * (NM_ * S_), n0);
    const int lane = threadIdx.x & 31;
    const int col = n0 + (lane & 15);
    const int mb  = m0 + (lane >> 4) * 8;
    const float* ab = after + (size_t)be * CHW_;
    float* ob = out + (size_t)be * CHW_;
#pragma unroll
    for (int r = 0; r < 8; ++r)
        ob[(mb + r) * S_ + col] = ab[(mb + r) * S_ + col] + acc[r] + b2[mb + r];
}

extern "C" void kernel_launch(void* const* d_in, const int* in_sizes, int n_in,
                              void* d_out, int out_size, void* d_ws, size_t ws_size,
                              hipStream_t stream) {
    const float* x    = (const float*)d_in[0];
    const float* ln_g = (const float*)d_in[1];
    const float* ln_b = (const float*)d_in[2];
    const float* Wv   = (const float*)d_in[3];
    const float* Wk   = (const float*)d_in[4];
    const float* Wq   = (const float*)d_in[5];
    const float* Wout = (const float*)d_in[6];
    const float* Wm1  = (const float*)d_in[7];
    const float* bm1  = (const float*)d_in[8];
    const float* Wm2  = (const float*)d_in[9];
    const float* bm2  = (const float*)d_in[10];
    float* out = (float*)d_out;

    float* ws = (float*)d_ws;
    float* pre   = ws;                       // [B,E,C,S]   (reused as transformed)
    float* vbuf  = ws + (size_t)NX_;         // [B,E,NH,S]
    float* kbuf  = ws + 2 * (size_t)NX_;     // [B,E,NH,S]  (reused as after)
    float* qbuf  = ws + 3 * (size_t)NX_;     // [B,E,NH,S]  (reused as h1)
    float* gram  = ws + 4 * (size_t)NX_;     // [B,NH,E,E]  640000 floats
    float* vmean = gram + (size_t)B_ * NH_ * E_ * E_;  // [B,NH,S] 524288 floats

    // 1) LayerNorm over [C,H,W] per (b,e)
    ln_kernel<<<B_ * E_, 256, 0, stream>>>(x, ln_g, ln_b, pre);

    // 2) v,k,q = W_{v,k,q} @ pre   (shared B-operand)
    qkv_kernel<<<12800, 256, 0, stream>>>(pre, Wv, Wk, Wq, vbuf, kbuf, qbuf);

    // 3) gram[b,h,i,j] then softmax over i
    gram_kernel<<<512, 256, 0, stream>>>(kbuf, qbuf, gram);
    softmax_kernel<<<B_ * NH_, 64, 0, stream>>>(gram);

    // 4) ensemble mean of v
    vmean_kernel<<<(B_ * NH_ * S_) / 256, 256, 0, stream>>>(vbuf, vmean);

    // 5) transformed = v + weights^T (v - mean)   -> overwrite `pre` buffer
    mix_kernel<<<16384, 256, 0, stream>>>(vbuf, vmean, gram, pre);

    // 6) after = x + W_out @ transformed          -> overwrite kbuf
    wout_kernel<<<12800, 256, 0, stream>>>(pre, Wout, x, kbuf);

    // 7) h1 = gelu(W_m1 @ after + b_m1)           -> overwrite qbuf
    m1_kernel<<<12800, 256, 0, stream>>>(kbuf, Wm1, bm1, qbuf);

    // 8) out = after + W_m2 @ h1 + b_m2
    m2_kernel<<<12800, 256, 0, stream>>>(qbuf, Wm2, bm2, kbuf, out);
}